// GATLayer_17678085390351
// MI455X (gfx1250) — compile-verified
//
#include <hip/hip_runtime.h>
#include <hip/hip_bf16.h>
#include <math.h>

// Problem constants (match reference)
#define NN   32000
#define FIN  256
#define HH   4
#define OUTD 64
#define HO   (HH * OUTD)   // 256
#define EE   512000
#define NEG_SLOPE 0.2f

typedef float v2f __attribute__((ext_vector_type(2)));
typedef float v8f __attribute__((ext_vector_type(8)));

// ---------------------------------------------------------------------------
// Kernel 0: init output / emax / denom
// ---------------------------------------------------------------------------
__global__ void gat_init(float* __restrict__ out,
                         float* __restrict__ emax,
                         float* __restrict__ denom) {
    int i = blockIdx.x * blockDim.x + threadIdx.x;
    if (i < NN * HO) out[i] = 0.0f;
    if (i < NN * HH) {
        emax[i]  = -INFINITY;
        denom[i] = 0.0f;
    }
}

// ---------------------------------------------------------------------------
// Kernel 1: ft = feat @ W via V_WMMA_F32_16X16X4_F32 (fp32, matches reference).
// Each wave computes a 16x64 output strip: one shared A fragment feeds 4
// INDEPENDENT accumulators (4 col tiles), so the WMMA chains can overlap
// instead of serializing on the D->C dependency, and A traffic drops 4x.
//
// Layouts (ISA 7.12.2, 32-bit):
//   A (16x4):  lane m = lane&15; half-wave selects K pair; VGPR = K within pair
//   B (4x16):  lane n = lane&15; half-wave selects K pair; VGPR = K within pair
//   C/D:       VGPR v -> rows v (lanes 0-15) / v+8 (lanes 16-31); col = lane&15
// ---------------------------------------------------------------------------
__global__ void gat_gemm_wmma(const float* __restrict__ feat,
                              const float* __restrict__ W,
                              float* __restrict__ ft) {
    const int lane   = threadIdx.x & 31;
    const int waveId = threadIdx.x >> 5;
    const int tile   = blockIdx.x * 8 + waveId;     // 8 waves per block
    const int colQuads = HO / 64;                   // 4
    const int rowTile  = tile / colQuads;
    const int colQuad  = tile % colQuads;
    const int row0 = rowTile * 16;
    const int col0 = colQuad * 64;

    const int n  = lane & 15;
    const int kk = (lane >> 4) << 1;                // 0 or 2

    const float* __restrict__ arow = feat + (size_t)(row0 + n) * FIN + kk;
    const float* __restrict__ bbase = W   + (size_t)kk * HO + col0 + n;

    v8f acc0 = {}, acc1 = {}, acc2 = {}, acc3 = {};

#pragma unroll 4
    for (int k = 0; k < FIN; k += 4) {
        v2f a;
        a.x = arow[k];
        a.y = arow[k + 1];

        const float* bk = bbase + (size_t)k * HO;
        v2f b0, b1, b2, b3;
        b0.x = bk[0];       b0.y = bk[HO];
        b1.x = bk[16];      b1.y = bk[HO + 16];
        b2.x = bk[32];      b2.y = bk[HO + 32];
        b3.x = bk[48];      b3.y = bk[HO + 48];

        acc0 = __builtin_amdgcn_wmma_f32_16x16x4_f32(false, a, false, b0,
                                                     (short)0, acc0, false, false);
        acc1 = __builtin_amdgcn_wmma_f32_16x16x4_f32(false, a, false, b1,
                                                     (short)0, acc1, false, false);
        acc2 = __builtin_amdgcn_wmma_f32_16x16x4_f32(false, a, false, b2,
                                                     (short)0, acc2, false, false);
        acc3 = __builtin_amdgcn_wmma_f32_16x16x4_f32(false, a, false, b3,
                                                     (short)0, acc3, false, false);
    }

    const int rowOff = (lane >> 4) << 3;            // 0 or 8
    float* __restrict__ obase = ft + (size_t)(row0 + rowOff) * HO + col0 + n;
#pragma unroll
    for (int v = 0; v < 8; ++v) {
        float* orow = obase + (size_t)v * HO;
        orow[0]  = acc0[v];
        orow[16] = acc1[v];
        orow[32] = acc2[v];
        orow[48] = acc3[v];
    }
}

// ---------------------------------------------------------------------------
// Kernel 2: per-(node, head) attention logits; one wave per (n,h)
// ---------------------------------------------------------------------------
__global__ void gat_logits(const float* __restrict__ ft,
                           const float* __restrict__ w_att_src,
                           const float* __restrict__ w_att_dst,
                           float* __restrict__ e_src,
                           float* __restrict__ e_dst) {
    int warp = (blockIdx.x * blockDim.x + threadIdx.x) >> 5;
    int lane = threadIdx.x & 31;
    if (warp >= NN * HH) return;
    int n = warp >> 2;        // / HH
    int h = warp & 3;         // % HH

    const float* row = ft + (size_t)n * HO + h * OUTD;
    const float* ws  = w_att_src + h * OUTD;
    const float* wd  = w_att_dst + h * OUTD;

    float s = row[lane] * ws[lane] + row[lane + 32] * ws[lane + 32];
    float d = row[lane] * wd[lane] + row[lane + 32] * wd[lane + 32];
#pragma unroll
    for (int off = 16; off > 0; off >>= 1) {
        s += __shfl_down(s, off, 32);
        d += __shfl_down(d, off, 32);
    }
    if (lane == 0) {
        e_src[warp] = s;
        e_dst[warp] = d;
    }
}

// ---------------------------------------------------------------------------
// Sign-aware float atomic max (int-compare trick, init must be -inf)
// ---------------------------------------------------------------------------
__device__ __forceinline__ void atomicMaxFloat(float* addr, float value) {
    if (value >= 0.0f)
        atomicMax((int*)addr, __float_as_int(value));
    else
        atomicMin((unsigned int*)addr, __float_as_uint(value));
}

// ---------------------------------------------------------------------------
// Kernel 3: edge logits + leaky relu + segment max
// ---------------------------------------------------------------------------
__global__ void gat_edge_max(const int* __restrict__ src,
                             const int* __restrict__ dst,
                             const float* __restrict__ e_src,
                             const float* __restrict__ e_dst,
                             float* __restrict__ e_edge,
                             float* __restrict__ emax) {
    int i = blockIdx.x * blockDim.x + threadIdx.x;
    if (i >= EE * HH) return;
    int e = i >> 2;
    int h = i & 3;
    float v = e_src[src[e] * HH + h] + e_dst[dst[e] * HH + h];
    v = (v >= 0.0f) ? v : NEG_SLOPE * v;
    e_edge[i] = v;
    atomicMaxFloat(&emax[dst[e] * HH + h], v);
}

// ---------------------------------------------------------------------------
// Kernel 4: exp(e - max) + segment sum of denominators
// ---------------------------------------------------------------------------
__global__ void gat_edge_exp(const int* __restrict__ dst,
                             const float* __restrict__ emax,
                             float* __restrict__ e_edge,
                             float* __restrict__ denom) {
    int i = blockIdx.x * blockDim.x + threadIdx.x;
    if (i >= EE * HH) return;
    int e = i >> 2;
    int h = i & 3;
    float ex = __expf(e_edge[i] - emax[dst[e] * HH + h]);
    e_edge[i] = ex;
    atomicAdd(&denom[dst[e] * HH + h], ex);
}

// ---------------------------------------------------------------------------
// Kernel 5: message + scatter-add aggregation. One 256-thread block per edge:
// thread c handles feature column c (head = c/64). ft[src] row read is fully
// coalesced (1 KB); output uses global float atomics into out[dst].
// ---------------------------------------------------------------------------
__global__ void gat_aggregate(const int* __restrict__ src,
                              const int* __restrict__ dst,
                              const float* __restrict__ dist,
                              const float* __restrict__ ft,
                              const float* __restrict__ e_edge,
                              const float* __restrict__ denom,
                              float* __restrict__ out) {
    int e = blockIdx.x;
    int c = threadIdx.x;
    int h = c >> 6;
    int s = src[e];
    int d = dst[e];
    float a = (e_edge[e * HH + h] / denom[d * HH + h]) * dist[e];
    atomicAdd(&out[(size_t)d * HO + c], ft[(size_t)s * HO + c] * a);
}

// ---------------------------------------------------------------------------
// Launch
// ---------------------------------------------------------------------------
extern "C" void kernel_launch(void* const* d_in, const int* in_sizes, int n_in,
                              void* d_out, int out_size, void* d_ws, size_t ws_size,
                              hipStream_t stream) {
    const float* feat      = (const float*)d_in[0];   // [N, FIN]
    const float* dist      = (const float*)d_in[1];   // [E, 1, 1]
    const float* W         = (const float*)d_in[2];   // [FIN, H*OUT]
    const float* w_att_src = (const float*)d_in[3];   // [1, H, OUT]
    const float* w_att_dst = (const float*)d_in[4];   // [1, H, OUT]
    const int*   src       = (const int*)d_in[5];     // [E]
    const int*   dst       = (const int*)d_in[6];     // [E]
    float*       out       = (float*)d_out;           // [N, H*OUT]

    // Workspace layout (bytes): ~43.0 MB total
    char* ws = (char*)d_ws;
    float* ft     = (float*)ws;                          ws += (size_t)NN * HO * 4;   // 32.768 MB
    float* e_src  = (float*)ws;                          ws += (size_t)NN * HH * 4;   // 0.512 MB
    float* e_dst  = (float*)ws;                          ws += (size_t)NN * HH * 4;
    float* emax   = (float*)ws;                          ws += (size_t)NN * HH * 4;
    float* denom  = (float*)ws;                          ws += (size_t)NN * HH * 4;
    float* e_edge = (float*)ws;                          ws += (size_t)EE * HH * 4;   // 8.192 MB

    // 0) init out / emax / denom
    {
        int total = NN * HO;                              // covers N*H too
        gat_init<<<(total + 255) / 256, 256, 0, stream>>>(out, emax, denom);
    }
    // 1) GEMM: 2000 row-tiles * 4 col-quads = 8000 wave-strips, 8 waves/block
    {
        int strips = (NN / 16) * (HO / 64);
        gat_gemm_wmma<<<strips / 8, 256, 0, stream>>>(feat, W, ft);
    }
    // 2) node logits: one wave per (n,h)
    {
        int warps = NN * HH;
        gat_logits<<<(warps * 32 + 255) / 256, 256, 0, stream>>>(ft, w_att_src, w_att_dst,
                                                                 e_src, e_dst);
    }
    // 3) edge max
    {
        int total = EE * HH;
        gat_edge_max<<<(total + 255) / 256, 256, 0, stream>>>(src, dst, e_src, e_dst,
                                                              e_edge, emax);
    }
    // 4) edge exp + denom
    {
        int total = EE * HH;
        gat_edge_exp<<<(total + 255) / 256, 256, 0, stream>>>(dst, emax, e_edge, denom);
    }
    // 5) aggregate: one block per edge
    {
        gat_aggregate<<<EE, HO, 0, stream>>>(src, dst, dist, ft, e_edge, denom, out);
    }
}